// Attention_54717883351680
// MI455X (gfx1250) — compile-verified
//
#include <hip/hip_runtime.h>

// result[i, j, c] = (X @ M)[i, c] + b_out[c]   for all j in [0,2048)
//   M[d][c] = W_qkv[2][d] * W_out[c][0]   (16x16)
// Store-bandwidth-bound: 256 MB out, ~11 us floor at 23.3 TB/s.

typedef __attribute__((ext_vector_type(2))) float v2f;
typedef __attribute__((ext_vector_type(4))) float v4f;
typedef __attribute__((ext_vector_type(8))) float v8f;

#define SEQ_L 2048
#define JC    128   // j columns handled per block; grid.y = SEQ_L/JC = 16

__global__ __launch_bounds__(256) void attn_bcast_wmma_kernel(
    const float* __restrict__ x,      // (2048, 16)
    const float* __restrict__ Wqkv,   // (3, 16) row-major; row 2 = v-projection
    const float* __restrict__ Wout,   // (16, 1)
    const float* __restrict__ bout,   // (16,)
    float* __restrict__ out)          // (2048, 2048, 16)
{
    __shared__ float rowvec[16 * 16];   // R tile: 16 rows x 16 cols

    const int tid  = threadIdx.x;
    const int tile = blockIdx.x;        // which 16-row tile of i
    const int jch  = blockIdx.y;        // which 128-wide j chunk

    // ---- Wave 0: R_tile = X_tile(16x16) @ M(16x16) + b  via 4x V_WMMA_F32_16X16X4_F32
    if (tid < 32) {                     // wave-uniform branch -> EXEC all 1s in wave 0
        const int lane = tid;
        const int rc   = lane & 15;     // A: row;  B/C/D: column
        const int hi   = lane >> 4;     // lane-half selects K offset +2 (A/B f32 layout)

        const float wcol = Wout[rc];    // W_out[col]
        const float bo   = bout[rc];

        v8f acc;
        #pragma unroll
        for (int r = 0; r < 8; ++r) acc[r] = bo;   // C[m][n] = b_out[n]

        const float* xrow = x + (size_t)(tile * 16 + rc) * 16;

        #pragma unroll
        for (int s = 0; s < 4; ++s) {
            const int k = s * 4 + hi * 2;
            v2f a, b;
            // A 16x4 f32: v0 = A[row][k], v1 = A[row][k+1] (lane-half -> k+2,k+3)
            a.x = xrow[k];
            a.y = xrow[k + 1];
            // B 4x16 f32 (mirrored layout): v0 = M[k][col], v1 = M[k+1][col]
            b.x = Wqkv[32 + k]     * wcol;
            b.y = Wqkv[32 + k + 1] * wcol;
            acc = __builtin_amdgcn_wmma_f32_16x16x4_f32(
                /*neg_a=*/false, a, /*neg_b=*/false, b,
                /*c_mod=*/(short)0, acc, /*reuse_a=*/false, /*reuse_b=*/false);
        }

        // D layout: VGPR r holds row (r + 8*hi), lane&15 = col
        #pragma unroll
        for (int r = 0; r < 8; ++r) {
            rowvec[(r + hi * 8) * 16 + rc] = acc[r];
        }
    }
    __syncthreads();

    // ---- Streaming phase: replicate each 64B row vector across its j range.
    // 256 threads: (i_local 0..15) x (quad 0..3) x (jsub 0..3); each 16-lane
    // group covers a contiguous 256B span -> coalesced global_store_b128.
    const int i_local = tid >> 4;
    const int q       = (tid >> 2) & 3;
    const int jsub    = tid & 3;

    const v4f val = *(const v4f*)&rowvec[i_local * 16 + q * 4];

    const int   i_glob = tile * 16 + i_local;
    const int   j0     = jch * JC + jsub;
    float* op = out + (((size_t)i_glob * SEQ_L + j0) * 16 + q * 4);

    // Output (256MB) >> L2 (192MB) and is write-once: non-temporal stores.
    #pragma unroll 8
    for (int t = 0; t < JC / 4; ++t) {
        __builtin_nontemporal_store(val, (v4f*)op);
        op += 4 * 16;   // advance j by 4 cells (64 floats)
    }
}

extern "C" void kernel_launch(void* const* d_in, const int* in_sizes, int n_in,
                              void* d_out, int out_size, void* d_ws, size_t ws_size,
                              hipStream_t stream) {
    const float* x    = (const float*)d_in[0];  // (1,2048,16)
    const float* Wqkv = (const float*)d_in[1];  // (3,16)
    const float* Wout = (const float*)d_in[2];  // (16,1)
    const float* bout = (const float*)d_in[3];  // (16,)
    float* out = (float*)d_out;                 // (1,2048,2048,16)

    dim3 grid(SEQ_L / 16, SEQ_L / JC, 1);       // (128, 16)
    dim3 block(256, 1, 1);
    attn_bcast_wmma_kernel<<<grid, block, 0, stream>>>(x, Wqkv, Wout, bout, out);
}